// AudioClassifier_18004502905141
// MI455X (gfx1250) — compile-verified
//
#include <hip/hip_runtime.h>
#include <hip/hip_bf16.h>

typedef __attribute__((ext_vector_type(2))) float v2f;
typedef __attribute__((ext_vector_type(8))) float v8f;

#define HS 64
#define NCLS 527
#define TSTEPS 1024

__device__ __forceinline__ float leakyf(float v) { return v >= 0.0f ? v : 0.2f * v; }
__device__ __forceinline__ float sigmf(float x) { return 1.0f / (1.0f + __expf(-x)); }
__device__ __forceinline__ float tanh_fast(float x) {
    x = fminf(fmaxf(x, -15.0f), 15.0f);
    float e = __expf(2.0f * x);
    return (e - 1.0f) / (e + 1.0f);
}

// ---------------------------------------------------------------------------
// Kernel 1: conv pyramid over the only samples that matter (x[b,0,0:64])
// -> xt[b, 0:64] -> gi[b, 0:192] = xt @ w_ih^T + b_ih   (written to d_ws)
// One block per batch, 64 threads.
// ---------------------------------------------------------------------------
__global__ void __launch_bounds__(64)
conv_head_kernel(const float* __restrict__ x,
                 const float* __restrict__ w0, const float* __restrict__ b0,
                 const float* __restrict__ w1, const float* __restrict__ b1,
                 const float* __restrict__ w2, const float* __restrict__ b2,
                 const float* __restrict__ w3, const float* __restrict__ b3,
                 const float* __restrict__ w4, const float* __restrict__ b4,
                 const float* __restrict__ w5, const float* __restrict__ b5,
                 const float* __restrict__ w_ih, const float* __restrict__ b_ih,
                 float* __restrict__ gi_g)
{
    const int b = blockIdx.x;
    const int t = threadIdx.x;
    __shared__ float xin[64];
    __shared__ float y0[16 * 32];
    __shared__ float y1[16 * 16];
    __shared__ float y2[32 * 8];
    __shared__ float y3[32 * 4];
    __shared__ float y4[64 * 2];
    __shared__ float y5[64];

    xin[t] = x[(size_t)b * 65536 + t];
    __syncthreads();

    // layer 0: 1->16 ch, 32 out positions
    for (int idx = t; idx < 16 * 32; idx += 64) {
        int c = idx >> 5, p = idx & 31;
        float acc = b0[c];
        #pragma unroll
        for (int k = 0; k < 3; ++k) {
            int pos = 2 * p - 1 + k;
            if (pos >= 0) acc += w0[c * 3 + k] * xin[pos];
        }
        y0[idx] = leakyf(acc);
    }
    __syncthreads();
    // layer 1: 16->16, 16 positions
    for (int idx = t; idx < 16 * 16; idx += 64) {
        int c = idx >> 4, p = idx & 15;
        float acc = b1[c];
        for (int i = 0; i < 16; ++i)
            #pragma unroll
            for (int k = 0; k < 3; ++k) {
                int pos = 2 * p - 1 + k;
                if (pos >= 0) acc += w1[(c * 16 + i) * 3 + k] * y0[i * 32 + pos];
            }
        y1[idx] = leakyf(acc);
    }
    __syncthreads();
    // layer 2: 16->32, 8 positions
    for (int idx = t; idx < 32 * 8; idx += 64) {
        int c = idx >> 3, p = idx & 7;
        float acc = b2[c];
        for (int i = 0; i < 16; ++i)
            #pragma unroll
            for (int k = 0; k < 3; ++k) {
                int pos = 2 * p - 1 + k;
                if (pos >= 0) acc += w2[(c * 16 + i) * 3 + k] * y1[i * 16 + pos];
            }
        y2[idx] = leakyf(acc);
    }
    __syncthreads();
    // layer 3: 32->32, 4 positions
    for (int idx = t; idx < 32 * 4; idx += 64) {
        int c = idx >> 2, p = idx & 3;
        float acc = b3[c];
        for (int i = 0; i < 32; ++i)
            #pragma unroll
            for (int k = 0; k < 3; ++k) {
                int pos = 2 * p - 1 + k;
                if (pos >= 0) acc += w3[(c * 32 + i) * 3 + k] * y2[i * 8 + pos];
            }
        y3[idx] = leakyf(acc);
    }
    __syncthreads();
    // layer 4: 32->64, 2 positions
    for (int idx = t; idx < 64 * 2; idx += 64) {
        int c = idx >> 1, p = idx & 1;
        float acc = b4[c];
        for (int i = 0; i < 32; ++i)
            #pragma unroll
            for (int k = 0; k < 3; ++k) {
                int pos = 2 * p - 1 + k;
                if (pos >= 0) acc += w4[(c * 32 + i) * 3 + k] * y3[i * 4 + pos];
            }
        y4[idx] = leakyf(acc);
    }
    __syncthreads();
    // layer 5: 64->64, 1 position -> xt
    {
        int c = t;
        float acc = b5[c];
        for (int i = 0; i < 64; ++i)
            #pragma unroll
            for (int k = 1; k < 3; ++k)   // pos = k-1 in {0,1}
                acc += w5[(c * 64 + i) * 3 + k] * y4[i * 2 + (k - 1)];
        y5[c] = leakyf(acc);
    }
    __syncthreads();
    // gi = xt @ w_ih^T + b_ih  (192 outputs)
    for (int j = t; j < 3 * HS; j += 64) {
        float acc = b_ih[j];
        const float* w = w_ih + j * HS;
        #pragma unroll 8
        for (int k = 0; k < HS; ++k) acc += w[k] * y5[k];
        gi_g[b * 3 * HS + j] = acc;
    }
}

// ---------------------------------------------------------------------------
// Kernel 2: 1024-step GRU scan with V_WMMA_F32_16X16X4_F32, then dense +
// log_softmax. Single workgroup, 24 waves (768 threads), h/gi/gh in LDS,
// w_hh^T tile resident in registers per wave for all 1024 steps.
// ---------------------------------------------------------------------------
__global__ void __launch_bounds__(768)
gru_scan_kernel(const float* __restrict__ h0,
                const float* __restrict__ w_hh, const float* __restrict__ b_hh,
                const float* __restrict__ w_dense, const float* __restrict__ b_dense,
                const float* __restrict__ gi_g,
                float* __restrict__ out)
{
    __shared__ float h_s[32 * HS];        // 8 KB
    __shared__ float gi_s[32 * 3 * HS];   // 24 KB
    __shared__ float gh_s[32 * 3 * HS];   // 24 KB

    const int tid  = threadIdx.x;
    const int lane = tid & 31;
    const int wv   = tid >> 5;            // 0..23
    const int mt   = wv & 1;              // M tile: rows mt*16..
    const int nt   = wv >> 1;             // N tile: cols nt*16.. (0..11)
    const int nl   = lane & 15;
    const int kh   = lane >> 4;           // K half: lanes 0-15 -> K{0,1}, 16-31 -> K{2,3}
    const int ncol = nt * 16 + nl;

    for (int i = tid; i < 32 * HS; i += 768) h_s[i] = h0[i];
    for (int i = tid; i < 32 * 3 * HS; i += 768) gi_s[i] = gi_g[i];

    // Preload this wave's B slab of w_hh^T: B[k4+kk][n] = w_hh[ncol*64 + k4+kk]
    v2f breg[16];
    #pragma unroll
    for (int k = 0; k < 16; ++k) {
        const float* p = w_hh + ncol * HS + k * 4 + kh * 2;
        breg[k] = v2f{p[0], p[1]};
    }
    const float bhh = b_hh[ncol];
    const int arow = mt * 16 + nl;
    __syncthreads();

    for (int step = 0; step < TSTEPS; ++step) {
        // gh tile = h(16x64) x w_hh^T(64x16) + b_hh, K split over 4 accumulators
        v8f c0, c1, c2, c3;
        #pragma unroll
        for (int v = 0; v < 8; ++v) { c0[v] = bhh; c1[v] = 0.f; c2[v] = 0.f; c3[v] = 0.f; }
        #pragma unroll
        for (int k = 0; k < 16; k += 4) {
            const float* hp0 = &h_s[arow * HS + (k + 0) * 4 + kh * 2];
            const float* hp1 = &h_s[arow * HS + (k + 1) * 4 + kh * 2];
            const float* hp2 = &h_s[arow * HS + (k + 2) * 4 + kh * 2];
            const float* hp3 = &h_s[arow * HS + (k + 3) * 4 + kh * 2];
            v2f a0 = v2f{hp0[0], hp0[1]};
            v2f a1 = v2f{hp1[0], hp1[1]};
            v2f a2 = v2f{hp2[0], hp2[1]};
            v2f a3 = v2f{hp3[0], hp3[1]};
            c0 = __builtin_amdgcn_wmma_f32_16x16x4_f32(false, a0, false, breg[k + 0], (short)0, c0, false, false);
            c1 = __builtin_amdgcn_wmma_f32_16x16x4_f32(false, a1, false, breg[k + 1], (short)0, c1, false, false);
            c2 = __builtin_amdgcn_wmma_f32_16x16x4_f32(false, a2, false, breg[k + 2], (short)0, c2, false, false);
            c3 = __builtin_amdgcn_wmma_f32_16x16x4_f32(false, a3, false, breg[k + 3], (short)0, c3, false, false);
        }
        // C/D layout: VGPR v -> (M = v + kh*8, N = ncol)
        #pragma unroll
        for (int v = 0; v < 8; ++v)
            gh_s[(mt * 16 + kh * 8 + v) * (3 * HS) + ncol] = c0[v] + c1[v] + c2[v] + c3[v];
        __syncthreads();

        // elementwise GRU gates: 32*64 elements over 768 threads
        for (int i = tid; i < 32 * HS; i += 768) {
            int b = i >> 6, j = i & 63;
            const float* gir = &gi_s[b * 3 * HS];
            const float* ghr = &gh_s[b * 3 * HS];
            float r = sigmf(gir[j] + ghr[j]);
            float z = sigmf(gir[HS + j] + ghr[HS + j]);
            float n = tanh_fast(gir[2 * HS + j] + r * ghr[2 * HS + j]);
            h_s[i] = (1.0f - z) * n + z * h_s[i];
        }
        __syncthreads();
    }

    // dense (32x527) + log_softmax, one wave per batch row, logits in registers
    for (int b = wv; b < 32; b += 24) {
        float lv[17];
        float mx = -3.0e38f;
        int cnt = 0;
        for (int c = lane; c < NCLS; c += 32) {
            float acc = b_dense[c];
            const float* wd = w_dense + c * HS;
            const float* hb = &h_s[b * HS];
            #pragma unroll 8
            for (int k = 0; k < HS; ++k) acc += wd[k] * hb[k];
            lv[cnt++] = acc;
            mx = fmaxf(mx, acc);
        }
        #pragma unroll
        for (int off = 16; off >= 1; off >>= 1) mx = fmaxf(mx, __shfl_xor(mx, off, 32));
        float s = 0.0f;
        for (int i = 0; i < cnt; ++i) s += __expf(lv[i] - mx);
        #pragma unroll
        for (int off = 16; off >= 1; off >>= 1) s += __shfl_xor(s, off, 32);
        const float lse = __logf(s) + mx;
        cnt = 0;
        for (int c = lane; c < NCLS; c += 32) out[b * NCLS + c] = lv[cnt++] - lse;
    }
}

extern "C" void kernel_launch(void* const* d_in, const int* in_sizes, int n_in,
                              void* d_out, int out_size, void* d_ws, size_t ws_size,
                              hipStream_t stream) {
    const float* x       = (const float*)d_in[0];
    const float* h0      = (const float*)d_in[1];
    const float* w0      = (const float*)d_in[2];
    const float* b0      = (const float*)d_in[3];
    const float* w1      = (const float*)d_in[4];
    const float* b1      = (const float*)d_in[5];
    const float* w2      = (const float*)d_in[6];
    const float* b2      = (const float*)d_in[7];
    const float* w3      = (const float*)d_in[8];
    const float* b3      = (const float*)d_in[9];
    const float* w4      = (const float*)d_in[10];
    const float* b4      = (const float*)d_in[11];
    const float* w5      = (const float*)d_in[12];
    const float* b5      = (const float*)d_in[13];
    const float* w_ih    = (const float*)d_in[14];
    const float* w_hh    = (const float*)d_in[15];
    const float* b_ih    = (const float*)d_in[16];
    const float* b_hh    = (const float*)d_in[17];
    const float* w_dense = (const float*)d_in[18];
    const float* b_dense = (const float*)d_in[19];

    float* gi_g = (float*)d_ws;           // 32*192 floats = 24 KB scratch
    float* out  = (float*)d_out;

    conv_head_kernel<<<32, 64, 0, stream>>>(x, w0, b0, w1, b1, w2, b2, w3, b3,
                                            w4, b4, w5, b5, w_ih, b_ih, gi_g);
    gru_scan_kernel<<<1, 768, 0, stream>>>(h0, w_hh, b_hh, w_dense, b_dense,
                                           gi_g, out);
}